// MixLoraGate_10015863734801
// MI455X (gfx1250) — compile-verified
//
#include <hip/hip_runtime.h>
#include <math.h>

#define N_TOKENS    32768
#define DIM         2048
#define NUM_EXPERTS 8

typedef __attribute__((ext_vector_type(2))) float v2f;
typedef __attribute__((ext_vector_type(8))) float v8f;

// One wave handles 16 tokens. 8 waves / block -> 128 tokens / block.
// grid = 32768 / 128 = 256 blocks.
__global__ __launch_bounds__(256) void moe_gate_topk_kernel(
    const float* __restrict__ x,       // [N_TOKENS, DIM]
    const float* __restrict__ gate_w,  // [NUM_EXPERTS, DIM]
    float* __restrict__ out)           // [2*N] weights, then [2*N] indices (as float)
{
    // 8 waves * 16 tokens * 8 experts of logit scratch = 4 KB LDS
    __shared__ float lds_scr[8 * 16 * 8];

    const int tid  = threadIdx.x;     // 0..255
    const int wave = tid >> 5;        // 0..7
    const int lane = tid & 31;
    const int r    = lane & 15;       // A: row M, B: column N (expert)
    const int g    = lane >> 4;       // lane half -> hardware K slots {0,1} vs {2,3}

    const int tokBase = blockIdx.x * 128 + wave * 16;

    // x stream: lane loads x[tokBase + r][k + 4*g .. +3] each step
    const float* xrow = x + (size_t)(tokBase + r) * DIM + 4 * g;
    // gate_w: lanes r>=8 alias rows r&7; their products land in accumulator
    // columns 8..15 which are never read, so no masking is needed.
    const float* wrow = gate_w + (size_t)(r & 7) * DIM + 4 * g;

    v8f acc = {0.f, 0.f, 0.f, 0.f, 0.f, 0.f, 0.f, 0.f};

    #pragma unroll 8
    for (int k = 0; k < DIM; k += 8) {
        float4 xa = *(const float4*)(xrow + k);
        float4 wa = *(const float4*)(wrow + k);

        // WMMA 1: covers absolute K = {k, k+1} (g=0 lanes) and {k+4, k+5} (g=1 lanes)
        v2f a0 = {xa.x, xa.y};
        v2f b0 = {wa.x, wa.y};
        acc = __builtin_amdgcn_wmma_f32_16x16x4_f32(
                  false, a0, false, b0, (short)0, acc, false, false);

        // WMMA 2: covers absolute K = {k+2, k+3} and {k+6, k+7}
        v2f a1 = {xa.z, xa.w};
        v2f b1 = {wa.z, wa.w};
        acc = __builtin_amdgcn_wmma_f32_16x16x4_f32(
                  false, a1, false, b1, (short)0, acc, false, false);
    }

    // D layout: VGPR i -> lanes 0-15 hold (M=i, N=lane), lanes 16-31 hold (M=8+i, N=lane-16).
    // Spill valid expert columns (N = r < 8) to LDS so one lane per token can
    // read its 8 logits contiguously.
    float* scr = &lds_scr[wave * 128];
    if (r < 8) {
        #pragma unroll
        for (int i = 0; i < 8; ++i) {
            scr[(g * 8 + i) * 8 + r] = acc[i];
        }
    }
    __syncthreads();

    // Lanes 0..15: top-2 + softmax for token (tokBase + lane)
    if (lane < 16) {
        const float* v = &lds_scr[wave * 128 + lane * 8];
        float best = v[0];
        int   bi   = 0;
        float sec  = -INFINITY;
        int   si   = 0;
        #pragma unroll
        for (int e = 1; e < NUM_EXPERTS; ++e) {
            float val = v[e];
            if (val > best) {          // strict '>' keeps lowest index on ties (jax top_k)
                sec = best; si = bi;
                best = val; bi = e;
            } else if (val > sec) {
                sec = val; si = e;
            }
        }
        // softmax over the (descending) top-2: e0 = 1, e1 = exp(sec - best)
        float e1 = expf(sec - best);
        float s  = 1.0f + e1;
        float w0 = 1.0f / s;
        float w1 = e1 / s;

        const int gtok = tokBase + lane;
        out[2 * gtok + 0] = w0;
        out[2 * gtok + 1] = w1;
        float* oid = out + 2 * N_TOKENS;
        oid[2 * gtok + 0] = (float)bi;
        oid[2 * gtok + 1] = (float)si;
    }
}

extern "C" void kernel_launch(void* const* d_in, const int* in_sizes, int n_in,
                              void* d_out, int out_size, void* d_ws, size_t ws_size,
                              hipStream_t stream) {
    (void)in_sizes; (void)n_in; (void)out_size; (void)d_ws; (void)ws_size;
    const float* x      = (const float*)d_in[0];
    const float* gate_w = (const float*)d_in[1];
    float*       out    = (float*)d_out;

    dim3 grid(N_TOKENS / 128);  // 256 blocks
    dim3 block(256);            // 8 waves (wave32)
    moe_gate_topk_kernel<<<grid, block, 0, stream>>>(x, gate_w, out);
}